// EPrompt_68143951118593
// MI455X (gfx1250) — compile-verified
//
#include <hip/hip_runtime.h>
#include <hip/hip_bf16.h>

typedef __attribute__((ext_vector_type(2))) float v2f;
typedef __attribute__((ext_vector_type(4))) float v4f;
typedef __attribute__((ext_vector_type(8))) float v8f;

#define B_    256
#define S_    197
#define D_    768
#define POOL_ 50
#define DQ_   192          // D/4
#define PLEN_ (42 * 960)   // DUAL_LEN * PD = 40320 floats per prompt copy

// ---------------------------------------------------------------------------
// Kernel 1: x_mean[b, d] = mean_s x_embed[b, s, d]
// 155 MB streamed once -> non-temporal 128-bit loads, fully coalesced.
// ---------------------------------------------------------------------------
__global__ void mean_tokens_kernel(const float* __restrict__ x, float* __restrict__ xm) {
    const int b = blockIdx.x;
    const int t = threadIdx.x;                     // 0..191 (D/4 vec4 lanes)
    const v4f* p = reinterpret_cast<const v4f*>(x + (size_t)b * S_ * D_) + t;
    v4f acc = {0.f, 0.f, 0.f, 0.f};
    for (int s = 0; s < S_; ++s) {
        v4f v = __builtin_nontemporal_load(p + (size_t)s * (D_ / 4));
        acc += v;
    }
    const float inv = 1.0f / (float)S_;
    reinterpret_cast<v4f*>(xm + (size_t)b * D_)[t] = acc * inv;
}

// ---------------------------------------------------------------------------
// Kernel 2/3: Dst[M,N] = act(A[M,K] @ W[N,K]^T + bias[N]) via V_WMMA_F32_16X16X4_F32.
// One wave32 per 16x16 tile; 4 waves per block span 64 output columns.
// A-frag (16x4): lane l<16 holds A[tm+l, k0..k0+1]; lanes 16..31 hold k0+2..k0+3.
// B-frag (4x16): b = B[k0 + 2*half + j][tn + (l&15)] = W[tn + (l&15)][k0 + 2*half + j].
// C/D (16x16 f32): vgpr r, lane l -> row tm + 8*(l>>4) + r, col tn + (l&15).
// No divergence before WMMA -> EXEC all ones as the ISA requires.
// ---------------------------------------------------------------------------
template <bool RELU>
__global__ void wmma_gemm_bias_kernel(const float* __restrict__ A, const float* __restrict__ W,
                                      const float* __restrict__ bias, float* __restrict__ Dst,
                                      int M, int N, int K) {
    const int lane = threadIdx.x & 31;
    const int wave = threadIdx.x >> 5;
    const int tm   = blockIdx.x * 16;
    const int tn   = (blockIdx.y * 4 + wave) * 16;
    const int half = lane >> 4;     // K-pair selector
    const int l15  = lane & 15;     // M index (A) / N index (B)

    const float* arow = A + (size_t)(tm + l15) * K + half * 2;
    const float* wrow = W + (size_t)(tn + l15) * K + half * 2;

    v8f c = {};
#pragma unroll 4
    for (int k0 = 0; k0 < K; k0 += 4) {
        v2f a = *reinterpret_cast<const v2f*>(arow + k0);   // 8B-aligned (k0 % 4 == 0)
        v2f b = *reinterpret_cast<const v2f*>(wrow + k0);
        // 8 args: (neg_a, A, neg_b, B, c_mod, C, reuse_a, reuse_b)
        c = __builtin_amdgcn_wmma_f32_16x16x4_f32(false, a, false, b, (short)0, c, false, false);
    }

    const int   col = tn + l15;
    const float bv  = bias[col];
#pragma unroll
    for (int r = 0; r < 8; ++r) {
        float v = c[r] + bv;
        if (RELU) v = fmaxf(v, 0.f);
        Dst[(size_t)(tm + half * 8 + r) * N + col] = v;
    }
}

// ---------------------------------------------------------------------------
// Row-wise L2 normalize: out = in * rsqrt(max(sum(in^2), 1e-12)), C <= 256.
// ---------------------------------------------------------------------------
__global__ void l2norm_rows_kernel(const float* __restrict__ in, float* __restrict__ out, int C) {
    __shared__ float red[256];
    const int row = blockIdx.x;
    const int t   = threadIdx.x;
    float v = (t < C) ? in[(size_t)row * C + t] : 0.f;
    red[t] = v * v;
    __syncthreads();
    for (int s = 128; s > 0; s >>= 1) {
        if (t < s) red[t] += red[t + s];
        __syncthreads();
    }
    const float scale = rsqrtf(fmaxf(red[0], 1e-12f));
    if (t < C) out[(size_t)row * C + t] = v * scale;
}

// ---------------------------------------------------------------------------
// similarity[b, p] = dot(x_norm[b, :], key_norm[p, :]) over D/4 = 192.
// x_norm row staged in LDS; 50 lanes each do one 192-length dot. Tiny.
// ---------------------------------------------------------------------------
__global__ void similarity_kernel(const float* __restrict__ xn, const float* __restrict__ kn,
                                  float* __restrict__ sim) {
    __shared__ float xr[DQ_];
    const int b = blockIdx.x;
    const int t = threadIdx.x;                      // 64 threads
    for (int i = t; i < DQ_; i += 64) xr[i] = xn[(size_t)b * DQ_ + i];
    __syncthreads();
    if (t < POOL_) {
        const float* kr = kn + (size_t)t * DQ_;
        float acc = 0.f;
#pragma unroll 4
        for (int i = 0; i < DQ_; ++i) acc += xr[i] * kr[i];
        sim[(size_t)b * POOL_ + t] = acc;
    }
}

// ---------------------------------------------------------------------------
// batched_prompt: replicate prompt[layer_num] (40320 floats) 256x.
// 41 MB written once -> non-temporal 128-bit stores; src stays L2-hot.
// ---------------------------------------------------------------------------
__global__ void bcast_prompt_kernel(const float* __restrict__ prompt,
                                    const int* __restrict__ layer,
                                    float* __restrict__ out) {
    const int PLEN4 = PLEN_ / 4;   // 10080 vec4
    const v4f* src = reinterpret_cast<const v4f*>(prompt + (size_t)layer[0] * PLEN_);
    const int i = blockIdx.x * blockDim.x + threadIdx.x;
    if (i < PLEN4) {
        v4f v = src[i];
        __builtin_nontemporal_store(v, reinterpret_cast<v4f*>(out) + (size_t)blockIdx.y * PLEN4 + i);
    }
}

// ---------------------------------------------------------------------------
extern "C" void kernel_launch(void* const* d_in, const int* in_sizes, int n_in,
                              void* d_out, int out_size, void* d_ws, size_t ws_size,
                              hipStream_t stream) {
    const float* x_embed    = (const float*)d_in[0];
    const float* prompt     = (const float*)d_in[1];
    const float* prompt_key = (const float*)d_in[2];
    const float* W1         = (const float*)d_in[3];
    const float* b1         = (const float*)d_in[4];
    const float* W2         = (const float*)d_in[5];
    const float* b2         = (const float*)d_in[6];
    const int*   layer      = (const int*)d_in[7];

    float* out     = (float*)d_out;
    float* sim     = out;                         // [256, 50]
    float* x_norm  = out + B_ * POOL_;            // [256, 192]
    float* bprompt = x_norm + B_ * DQ_;           // [256, 42, 960]

    float* ws   = (float*)d_ws;
    float* xm   = ws;                 // [256, 768]
    float* h    = xm + B_ * D_;       // [256, 384]
    float* feat = h + B_ * (D_ / 2);  // [256, 192]
    float* keyn = feat + B_ * DQ_;    // [50, 192]

    // 1) token mean (the 155 MB stream — bandwidth floor of the whole op)
    mean_tokens_kernel<<<B_, D_ / 4, 0, stream>>>(x_embed, xm);

    // 2) h = relu(xm @ W1^T + b1): M=256, N=384, K=768 (WMMA f32 16x16x4)
    wmma_gemm_bias_kernel<true><<<dim3(B_ / 16, 384 / 64), 128, 0, stream>>>(
        xm, W1, b1, h, B_, 384, 768);

    // 3) feat = h @ W2^T + b2: M=256, N=192, K=384
    wmma_gemm_bias_kernel<false><<<dim3(B_ / 16, 192 / 64), 128, 0, stream>>>(
        h, W2, b2, feat, B_, 192, 384);

    // 4) x_norm (to output) and key_norm (to scratch)
    l2norm_rows_kernel<<<B_, 256, 0, stream>>>(feat, x_norm, DQ_);
    l2norm_rows_kernel<<<POOL_, 256, 0, stream>>>(prompt_key, keyn, DQ_);

    // 5) similarity = x_norm @ key_norm^T
    similarity_kernel<<<B_, 64, 0, stream>>>(x_norm, keyn, sim);

    // 6) batched prompt broadcast (41 MB NT store stream)
    bcast_prompt_kernel<<<dim3((PLEN_ / 4 + 255) / 256, B_), 256, 0, stream>>>(
        prompt, layer, bprompt);
}